// _GraphSAGELayer_75222057222467
// MI455X (gfx1250) — compile-verified
//
#include <hip/hip_runtime.h>
#include <hip/hip_bf16.h>

typedef float v2f __attribute__((ext_vector_type(2)));
typedef float v8f __attribute__((ext_vector_type(8)));

#define D 64  // D_IN == D_OUT == 64

// ---------------------------------------------------------------------------
// Kernel 1: zero the scratch (agg[N*64] followed by deg[N])
// ---------------------------------------------------------------------------
__global__ void _sage_zero_kernel(float* __restrict__ p, long long n) {
    long long i = (long long)blockIdx.x * blockDim.x + threadIdx.x;
    if (i < n) p[i] = 0.0f;
}

// ---------------------------------------------------------------------------
// Kernel 2: edge scatter.  16 threads per edge, 4 floats per thread.
// agg[dst] += x[src];  deg[dst] += 1
// ---------------------------------------------------------------------------
__global__ void _sage_scatter_kernel(const float* __restrict__ x,
                                     const int* __restrict__ ei,
                                     long long E,
                                     float* __restrict__ agg,
                                     float* __restrict__ deg) {
    long long t = (long long)blockIdx.x * blockDim.x + threadIdx.x;
    long long e = t >> 4;
    if (e >= E) return;
    int c   = (int)(t & 15);          // which float4 chunk of the 64-wide row
    int s   = ei[e];                  // src node
    int dst = ei[E + e];              // dst node
    const float4 v = ((const float4*)(x + (long long)s * D))[c];
    float* a = agg + (long long)dst * D + c * 4;
    atomicAdd(a + 0, v.x);
    atomicAdd(a + 1, v.y);
    atomicAdd(a + 2, v.z);
    atomicAdd(a + 3, v.w);
    if (c == 0) atomicAdd(&deg[dst], 1.0f);
}

// ---------------------------------------------------------------------------
// Kernel 3: fused dual-GEMM + bias + ReLU via V_WMMA_F32_16X16X4_F32.
// One wave32 computes a 16-row block: out[16x64] = relu(x@Ws + (agg/deg)@Wn + b)
// A layout (fp32 16x4): lane<16 holds {K=0,K=1}, lane>=16 holds {K=2,K=3}
// B layout (fp32 4x16): VGPR0 = row K=2*kh, VGPR1 = row K=2*kh+1, N = lane&15
// C/D layout: VGPR v -> row v + 8*(lane>=16), col = lane&15
// ---------------------------------------------------------------------------
__global__ __launch_bounds__(256) void _sage_gemm_kernel(
    const float* __restrict__ x, const float* __restrict__ agg,
    const float* __restrict__ deg,
    const float* __restrict__ Wself, const float* __restrict__ bself,
    const float* __restrict__ Wneigh, const float* __restrict__ bneigh,
    float* __restrict__ out, int nRowBlocks) {
    __shared__ float sWs[D * D];
    __shared__ float sWn[D * D];

    // Cooperative stage of both weight matrices into LDS (32 KB).
    {
        const float4* gs = (const float4*)Wself;
        const float4* gn = (const float4*)Wneigh;
        float4* ss = (float4*)sWs;
        float4* sn = (float4*)sWn;
        for (int i = threadIdx.x; i < (D * D) / 4; i += 256) {
            ss[i] = gs[i];
            sn[i] = gn[i];
        }
    }
    __syncthreads();

    const int wave = threadIdx.x >> 5;
    const int lane = threadIdx.x & 31;
    const int rb   = blockIdx.x * 8 + wave;
    if (rb >= nRowBlocks) return;           // whole-wave uniform exit

    const int row = lane & 15;              // M index for A, N index for B/C
    const int kh  = lane >> 4;              // K half select
    const long long grow = (long long)rb * 16 + row;

    const float* xr = x   + grow * D;
    const float* ar = agg + grow * D;
    const float  rdeg = 1.0f / fmaxf(deg[grow], 1.0f);

    v8f acc[4];
    acc[0] = v8f{}; acc[1] = v8f{}; acc[2] = v8f{}; acc[3] = v8f{};

    #pragma unroll
    for (int k = 0; k < 16; ++k) {          // K = 64 in steps of 4
        const int kb = k * 4 + kh * 2;
        v2f ax, ag;
        ax.x = xr[kb];
        ax.y = xr[kb + 1];
        ag.x = ar[kb] * rdeg;
        ag.y = ar[kb + 1] * rdeg;
        #pragma unroll
        for (int n = 0; n < 4; ++n) {       // 4 column tiles of 16
            const int c = n * 16 + row;
            v2f bs, bn;
            bs.x = sWs[kb * D + c];
            bs.y = sWs[(kb + 1) * D + c];
            bn.x = sWn[kb * D + c];
            bn.y = sWn[(kb + 1) * D + c];
            acc[n] = __builtin_amdgcn_wmma_f32_16x16x4_f32(
                false, ax, false, bs, (short)0, acc[n], false, false);
            acc[n] = __builtin_amdgcn_wmma_f32_16x16x4_f32(
                false, ag, false, bn, (short)0, acc[n], false, false);
        }
    }

    // bias + ReLU + store (D-matrix layout)
    #pragma unroll
    for (int n = 0; n < 4; ++n) {
        const int c = n * 16 + row;
        const float bsum = bself[c] + bneigh[c];
        #pragma unroll
        for (int v = 0; v < 8; ++v) {
            const long long orow = (long long)rb * 16 + v + kh * 8;
            float val = acc[n][v] + bsum;
            val = fmaxf(val, 0.0f);
            out[orow * D + c] = val;
        }
    }
}

// ---------------------------------------------------------------------------
// Kernel 4: scalar tail for row counts not divisible by 16 (not hit for
// N=100000 but kept for generality). One block per row, 64 threads.
// ---------------------------------------------------------------------------
__global__ void _sage_tail_kernel(const float* __restrict__ x,
                                  const float* __restrict__ agg,
                                  const float* __restrict__ deg,
                                  const float* __restrict__ Ws,
                                  const float* __restrict__ bs,
                                  const float* __restrict__ Wn,
                                  const float* __restrict__ bn,
                                  float* __restrict__ out,
                                  long long rowStart, long long N) {
    long long r = rowStart + blockIdx.x;
    if (r >= N) return;
    int c = threadIdx.x;
    float rd = 1.0f / fmaxf(deg[r], 1.0f);
    float a = bs[c] + bn[c];
    for (int k = 0; k < D; ++k)
        a += x[r * D + k] * Ws[k * D + c] + agg[r * D + k] * rd * Wn[k * D + c];
    out[r * D + c] = fmaxf(a, 0.0f);
}

// ---------------------------------------------------------------------------
extern "C" void kernel_launch(void* const* d_in, const int* in_sizes, int n_in,
                              void* d_out, int out_size, void* d_ws, size_t ws_size,
                              hipStream_t stream) {
    const float* x      = (const float*)d_in[0];
    const int*   ei     = (const int*)d_in[1];   // edge_index [2, E]
    const float* Wself  = (const float*)d_in[2];
    const float* bself  = (const float*)d_in[3];
    const float* Wneigh = (const float*)d_in[4];
    const float* bneigh = (const float*)d_in[5];
    float* out = (float*)d_out;

    const long long N = (long long)in_sizes[0] / D;
    const long long E = (long long)in_sizes[1] / 2;

    float* agg = (float*)d_ws;         // N*64 floats
    float* deg = agg + N * D;          // N floats

    // 1) zero agg + deg
    {
        long long tot = N * (D + 1);
        long long blocks = (tot + 255) / 256;
        _sage_zero_kernel<<<(unsigned)blocks, 256, 0, stream>>>(agg, tot);
    }
    // 2) scatter-add over edges
    {
        long long tot = E * 16;
        long long blocks = (tot + 255) / 256;
        _sage_scatter_kernel<<<(unsigned)blocks, 256, 0, stream>>>(x, ei, E, agg, deg);
    }
    // 3) fused WMMA dual-GEMM + bias + ReLU
    {
        long long nrb = N / 16;
        if (nrb > 0) {
            long long blocks = (nrb + 7) / 8;
            _sage_gemm_kernel<<<(unsigned)blocks, 256, 0, stream>>>(
                x, agg, deg, Wself, bself, Wneigh, bneigh, out, (int)nrb);
        }
        long long tailStart = nrb * 16;
        long long tail = N - tailStart;
        if (tail > 0) {
            _sage_tail_kernel<<<(unsigned)tail, D, 0, stream>>>(
                x, agg, deg, Wself, bself, Wneigh, bneigh, out, tailStart, N);
        }
    }
}